// InteractionAttention_75161927680557
// MI455X (gfx1250) — compile-verified
//
#include <hip/hip_runtime.h>
#include <hip/hip_bf16.h>

// ---------------------------------------------------------------------------
// InteractionAttention for MI455X (gfx1250, wave32, WMMA).
//
// Roofline: writing the attn output (12288x6144 f32 = 302MB) dominates
// (~13us at 23.3 TB/s). Batch ids are sorted -> mask is block-diagonal, so
// live GEMM work is only ~6 GF total: compute only diagonal blocks, keep
// everything fp32 via V_WMMA_F32_16X16X4_F32 (no precision tradeoff needed).
// ---------------------------------------------------------------------------

#define HIDDEN 256
#define NPROT  12288
#define NMOL   6144
#define NBATCH 32

typedef float v2f __attribute__((ext_vector_type(2)));
typedef float v8f __attribute__((ext_vector_type(8)));

// D = A(16x4 f32) x B(4x16 f32) + C(16x16 f32), wave32.
// A/B frag: lane m = lane&15 ; K pair = 2*(lane>=16) .. +1  (ISA 7.12.2)
// C/D frag: vgpr r holds row  m = r + 8*(lane>=16), col n = lane&15
__device__ __forceinline__ v8f wmma_f32(v2f a, v2f b, v8f c) {
  return __builtin_amdgcn_wmma_f32_16x16x4_f32(
      /*neg_a=*/false, a, /*neg_b=*/false, b,
      /*c_mod=*/(short)0, c, /*reuse_a=*/false, /*reuse_b=*/false);
}

// ---------------------------------------------------------------------------
// Per-batch mol column ranges: mstart[b] = first i with mol_batch[i] >= b.
// ---------------------------------------------------------------------------
__global__ void k_batch_starts(const int* __restrict__ mol_batch,
                               int* __restrict__ mstart) {
  int t = threadIdx.x;
  if (t <= NBATCH) {
    int lo = 0, hi = NMOL;
    while (lo < hi) {
      int mid = (lo + hi) >> 1;
      if (mol_batch[mid] < t) lo = mid + 1; else hi = mid;
    }
    mstart[t] = lo;
  }
}

// ---------------------------------------------------------------------------
// Zero-fill the attn output region (everything outside diagonal blocks is 0).
// ---------------------------------------------------------------------------
__global__ void k_zero(float4* __restrict__ p, size_t n4) {
  size_t i = (size_t)blockIdx.x * blockDim.x + threadIdx.x;
  size_t stride = (size_t)gridDim.x * blockDim.x;
  float4 z = make_float4(0.f, 0.f, 0.f, 0.f);
  for (; i < n4; i += stride) p[i] = z;
}

// ---------------------------------------------------------------------------
// out[n][h] = sum_k x[n][k]*W[h][k] + b[h]   (torch Linear).
// One wave per 16x16 output tile; 64 fp32 WMMA k-steps.
// ---------------------------------------------------------------------------
__global__ __launch_bounds__(32) void k_proj(const float* __restrict__ x,
                                             const float* __restrict__ W,
                                             const float* __restrict__ bias,
                                             float* __restrict__ out) {
  const int lane  = threadIdx.x;
  const int r0    = blockIdx.x * 16;
  const int h0    = blockIdx.y * 16;
  const int m     = lane & 15;              // A row / B col (same lane mapping)
  const int koff  = (lane >> 4) * 2;        // K pair selector
  const int mbase = (lane >> 4) * 8;        // C/D row base

  const float* __restrict__ xrow = x + (size_t)(r0 + m) * HIDDEN;
  const float* __restrict__ wrow = W + (size_t)(h0 + m) * HIDDEN; // B[k][n]=W[h0+n][k]

  v8f acc = {};
  for (int k = 0; k < HIDDEN; k += 4) {
    v2f a = *(const v2f*)(xrow + k + koff);
    v2f b = *(const v2f*)(wrow + k + koff);
    acc = wmma_f32(a, b, acc);
  }
  const int n = lane & 15;
  const float bv = bias[h0 + n];
  for (int r = 0; r < 8; ++r) {
    int mm = mbase + r;
    out[(size_t)(r0 + mm) * HIDDEN + h0 + n] = acc[r] + bv;
  }
}

// ---------------------------------------------------------------------------
// Block-diagonal attention. One 4-wave workgroup per 16 prot rows.
//   pass1: scaled scores -> attn (raw), via WMMA, waves split col tiles
//   scalar: row max / row sum (LDS reduced)
//   pass2: O += exp(s-max) @ V via WMMA; each wave owns 64 of 256 out cols
//   pass3: attn /= denom in place; out = O/denom (empty rows -> 0)
// ---------------------------------------------------------------------------
__global__ __launch_bounds__(128) void k_attn(
    const float* __restrict__ Q, const float* __restrict__ Km,
    const float* __restrict__ V, const int* __restrict__ prot_batch,
    const int* __restrict__ mstart, float* __restrict__ out,
    float* __restrict__ attn) {
  __shared__ int   ms[16], me[16];
  __shared__ float mx[16], dsum[16], red[128];

  const int tid  = threadIdx.x;
  const int wave = tid >> 5;
  const int lane = tid & 31;
  const int r0   = blockIdx.x * 16;

  if (tid < 16) {
    int b = prot_batch[r0 + tid];
    ms[tid] = mstart[b];
    me[tid] = mstart[b + 1];
  }
  __syncthreads();

  const int t0 = ms[0] >> 4;                 // batches sorted -> ms nondecreasing
  const int t1 = (me[15] + 15) >> 4;         // union col-tile range for these rows
  const int m     = lane & 15;
  const int koff  = (lane >> 4) * 2;
  const int mbase = (lane >> 4) * 8;
  const float scale = 0.0625f;               // 1/sqrt(256)

  // ---- pass 1: S = scale * Q K^T, write raw scaled scores into attn -------
  const float* __restrict__ qrow = Q + (size_t)(r0 + m) * HIDDEN;
  for (int ct = t0 + wave; ct < t1; ct += 4) {
    v8f acc = {};
    const float* __restrict__ krow = Km + (size_t)(ct * 16 + m) * HIDDEN; // B[k][n]=K[col][k]
    for (int k = 0; k < HIDDEN; k += 4) {
      v2f a = *(const v2f*)(qrow + k + koff);
      v2f b = *(const v2f*)(krow + k + koff);
      acc = wmma_f32(a, b, acc);
    }
    const int n = lane & 15;
    const int col = ct * 16 + n;
    for (int r = 0; r < 8; ++r) {
      int mm = mbase + r;
      if (col >= ms[mm] && col < me[mm])
        attn[(size_t)(r0 + mm) * NMOL + col] = acc[r] * scale;
    }
  }
  __syncthreads();

  // ---- row max / row sum (8 threads per row, LDS reduce) -------------------
  {
    const int r = tid >> 3, s = tid & 7;
    float lm = -__builtin_inff();
    for (int col = ms[r] + s; col < me[r]; col += 8)
      lm = fmaxf(lm, attn[(size_t)(r0 + r) * NMOL + col]);
    red[tid] = lm;
    __syncthreads();
    if (tid < 16) {
      float v = red[tid * 8];
      for (int i = 1; i < 8; ++i) v = fmaxf(v, red[tid * 8 + i]);
      mx[tid] = (me[tid] > ms[tid]) ? v : 0.f;
    }
    __syncthreads();
    float ls = 0.f;
    for (int col = ms[r] + s; col < me[r]; col += 8)
      ls += __expf(attn[(size_t)(r0 + r) * NMOL + col] - mx[r]);
    red[tid] = ls;
    __syncthreads();
    if (tid < 16) {
      float v = 0.f;
      for (int i = 0; i < 8; ++i) v += red[tid * 8 + i];
      dsum[tid] = v;
    }
    __syncthreads();
  }

  // ---- pass 2: O(16 x 64 per wave) += e @ V --------------------------------
  v8f o0 = {}, o1 = {}, o2 = {}, o3 = {};
  const int hb = wave * 64 + (lane & 15);
  for (int ct = t0; ct < t1; ++ct) {
    for (int ks = 0; ks < 4; ++ks) {
      const int kk   = ks * 4 + koff;        // local K pair within 16-col tile
      const int ccol = ct * 16 + kk;
      v2f s2 = *(const v2f*)(attn + (size_t)(r0 + m) * NMOL + ccol);
      v2f a;
      a.x = (ccol     >= ms[m] && ccol     < me[m]) ? __expf(s2.x - mx[m]) : 0.f;
      a.y = (ccol + 1 >= ms[m] && ccol + 1 < me[m]) ? __expf(s2.y - mx[m]) : 0.f;
      const float* __restrict__ vp = V + (size_t)ccol * HIDDEN + hb;
      v2f b;
      b.x = vp[0];           b.y = vp[HIDDEN];            o0 = wmma_f32(a, b, o0);
      b.x = vp[16];          b.y = vp[HIDDEN + 16];       o1 = wmma_f32(a, b, o1);
      b.x = vp[32];          b.y = vp[HIDDEN + 32];       o2 = wmma_f32(a, b, o2);
      b.x = vp[48];          b.y = vp[HIDDEN + 48];       o3 = wmma_f32(a, b, o3);
    }
  }

  // ---- write out = O / denom ----------------------------------------------
  {
    const int n = lane & 15;
    for (int r = 0; r < 8; ++r) {
      int mm = mbase + r;
      float inv = dsum[mm] > 0.f ? 1.f / dsum[mm] : 0.f;
      size_t base = (size_t)(r0 + mm) * HIDDEN + wave * 64 + n;
      out[base]      = o0[r] * inv;
      out[base + 16] = o1[r] * inv;
      out[base + 32] = o2[r] * inv;
      out[base + 48] = o3[r] * inv;
    }
  }
  __syncthreads();

  // ---- pass 3: normalize attn in place ------------------------------------
  {
    const int r = tid >> 3, s = tid & 7;
    const float inv = dsum[r] > 0.f ? 1.f / dsum[r] : 0.f;
    const float mxr = mx[r];
    for (int col = ms[r] + s; col < me[r]; col += 8) {
      size_t idx = (size_t)(r0 + r) * NMOL + col;
      attn[idx] = __expf(attn[idx] - mxr) * inv;
    }
  }
}

// ---------------------------------------------------------------------------
extern "C" void kernel_launch(void* const* d_in, const int* in_sizes, int n_in,
                              void* d_out, int out_size, void* d_ws, size_t ws_size,
                              hipStream_t stream) {
  const float* prot      = (const float*)d_in[0];
  const float* mol       = (const float*)d_in[1];
  const int*   protBatch = (const int*)d_in[2];
  const int*   molBatch  = (const int*)d_in[3];
  const float* Wq = (const float*)d_in[4];
  const float* bq = (const float*)d_in[5];
  const float* Wk = (const float*)d_in[6];
  const float* bk = (const float*)d_in[7];
  const float* Wv = (const float*)d_in[8];
  const float* bv = (const float*)d_in[9];

  float* out  = (float*)d_out;                       // [NPROT, HIDDEN]
  float* attn = out + (size_t)NPROT * HIDDEN;        // [NPROT, NMOL]

  // workspace: Q | K | V | mol batch starts
  float* Q = (float*)d_ws;
  float* K = Q + (size_t)NPROT * HIDDEN;
  float* V = K + (size_t)NMOL * HIDDEN;
  int*   mstart = (int*)(V + (size_t)NMOL * HIDDEN);

  k_batch_starts<<<1, 64, 0, stream>>>(molBatch, mstart);

  size_t n4 = (size_t)NPROT * NMOL / 4;
  k_zero<<<2048, 256, 0, stream>>>((float4*)attn, n4);

  dim3 gq(NPROT / 16, HIDDEN / 16);
  dim3 gm(NMOL  / 16, HIDDEN / 16);
  k_proj<<<gq, 32, 0, stream>>>(prot, Wq, bq, Q);
  k_proj<<<gm, 32, 0, stream>>>(mol,  Wk, bk, K);
  k_proj<<<gm, 32, 0, stream>>>(mol,  Wv, bv, V);

  k_attn<<<NPROT / 16, 128, 0, stream>>>(Q, K, V, protBatch, mstart, out, attn);
}